// QuantizedEmbedding_51393578664548
// MI455X (gfx1250) — compile-verified
//
#include <hip/hip_runtime.h>
#include <cstdint>
#include <cstddef>

// ---------------------------------------------------------------------------
// Quantized embedding gather + int4 dequant for MI455X (gfx1250).
//
// Roofline: ~392 MiB of traffic per launch (128 MiB gathered weight rows,
// 8 MiB gathered scales, 256 MiB streamed f32 output) -> ~17 us at 23.3 TB/s.
// VALU cost ~9 ops per packed int (~300M ops) is well under that, so the
// kernel is pure data movement. No matrix math => no WMMA; the CDNA5 path
// used is the async global->LDS DMA (global_load_async_to_lds_b128, tracked
// by ASYNCcnt) double-buffering the 8 KB weight rows, with th:TH_LOAD_NT on
// the use-once weight stream and non-temporal b128 output stores so the L2
// is reserved for the scales table.
// ---------------------------------------------------------------------------

#define QE_DIM        4096
#define QE_PACKED     2048   // int32 elements per weight row (each holds one int8)
#define QE_NGROUPS    128
#define QE_THREADS    256
#define QE_ROWS_PER_WG 4

typedef float v4f __attribute__((ext_vector_type(4)));
typedef int   v4i __attribute__((ext_vector_type(4)));

// Async DMA: each lane copies 16 bytes from global sbase+voff+IOFF to the LDS address in lds_off.
// GVS addressing: saddr = uniform row base (SGPR pair), vaddr = per-lane byte
// offset, ioffset = compile-time immediate. Tracked by ASYNCcnt. NT hint:
// weight rows are read exactly once -> do not retain in WGP$/L2.
template <int IOFF>
__device__ __forceinline__ void async_load_b128_nt(uint32_t lds_off,
                                                   uint32_t voff,
                                                   uint64_t sbase) {
  asm volatile("global_load_async_to_lds_b128 %0, %1, %2 offset:%3 th:TH_LOAD_NT"
               :: "v"(lds_off), "v"(voff), "s"(sbase), "i"(IOFF)
               : "memory");
}

template <int N>
__device__ __forceinline__ void wait_asynccnt() {
  asm volatile("s_wait_asynccnt %0" :: "i"(N) : "memory");
}

// Dequant 4 packed int8s (stored as int32) -> 8 f32 outputs.
// even = trunc(w/16) - 8  (float trunc is exact: w in [-128,127], /16 exact)
// odd  = (w & 15) - 8     (floor-mod, matches jnp.remainder with +16 divisor)
// Single rounding in the final multiply, bit-exact vs the reference.
__device__ __forceinline__ void dq4(v4i w, float s, v4f& lo, v4f& hi) {
  float r[8];
#pragma unroll
  for (int k = 0; k < 4; ++k) {
    int   wk = w[k];
    float ef = truncf((float)wk * 0.0625f) - 8.0f;   // high nibble, trunc-to-zero
    float of = (float)((wk & 15) - 8);               // low nibble, floor-mod
    r[2 * k]     = ef * s;
    r[2 * k + 1] = of * s;
  }
  lo = (v4f){r[0], r[1], r[2], r[3]};
  hi = (v4f){r[4], r[5], r[6], r[7]};
}

__global__ __launch_bounds__(QE_THREADS)
void qembed_dequant_kernel(const int* __restrict__ indices,
                           const int* __restrict__ weight,
                           const float* __restrict__ scales,
                           float* __restrict__ out,
                           int ntok) {
  // Double-buffered staging for one 8KB weight row per buffer (16 KB LDS).
  __shared__ alignas(16) int lds_buf[2][QE_PACKED];

  const int t    = (int)threadIdx.x;
  const int row0 = (int)blockIdx.x * QE_ROWS_PER_WG;
  if (row0 >= ntok) return;
  const int nr = min(QE_ROWS_PER_WG, ntok - row0);

  // Low 32 bits of a flat pointer into LDS == LDS byte offset (ISA 10.2).
  const uint32_t lds0 = (uint32_t)(uintptr_t)&lds_buf[0][0];
  const uint32_t voff = 16u * (uint32_t)t;  // chunk A byte offset within row

  // Prologue: kick off row 0 into buffer 0. Each lane stages exactly the
  // bytes it will later read, so no workgroup barrier is required; ASYNCcnt
  // waits (per-wave, in-order completion) are sufficient.
  int idx_next = indices[row0];
  {
    uint64_t base = (uint64_t)(uintptr_t)(weight + (size_t)idx_next * QE_PACKED);
    async_load_b128_nt<0>(lds0 + voff, voff, base);
    async_load_b128_nt<4096>(lds0 + 4096u + voff, voff, base);
  }

  for (int r = 0; r < nr; ++r) {
    const int      idx = idx_next;
    const uint32_t cur = (uint32_t)(r & 1);

    if (r + 1 < nr) {
      // Prefetch next row into the other buffer, then wait for current row
      // (<=2 outstanding = the pair just issued; the current pair landed).
      idx_next = indices[row0 + r + 1];
      uint64_t base =
          (uint64_t)(uintptr_t)(weight + (size_t)idx_next * QE_PACKED);
      uint32_t nb = lds0 + ((cur ^ 1u) * (uint32_t)(QE_PACKED * 4));
      async_load_b128_nt<0>(nb + voff, voff, base);
      async_load_b128_nt<4096>(nb + 4096u + voff, voff, base);
      wait_asynccnt<2>();
    } else {
      wait_asynccnt<0>();
    }

    // Outputs [8t,8t+8) and [2048+8t,+8) each sit in one scale group.
    const float sA = scales[(size_t)idx * QE_NGROUPS + (t >> 2)];
    const float sB = scales[(size_t)idx * QE_NGROUPS + 64 + (t >> 2)];

    const v4i* lp = (const v4i*)&lds_buf[cur][0];
    v4i wa = lp[t];        // ints [4t, 4t+4)
    v4i wb = lp[256 + t];  // ints [1024+4t, 1024+4t+4)

    v4f a0, a1, b0, b1;
    dq4(wa, sA, a0, a1);
    dq4(wb, sB, b0, b1);

    // 256 MiB streamed output: non-temporal b128 stores, fully coalesced.
    float* orow = out + (size_t)(row0 + r) * QE_DIM;
    __builtin_nontemporal_store(a0, (v4f*)(orow + 8 * t));
    __builtin_nontemporal_store(a1, (v4f*)(orow + 8 * t + 4));
    __builtin_nontemporal_store(b0, (v4f*)(orow + 2048 + 8 * t));
    __builtin_nontemporal_store(b1, (v4f*)(orow + 2048 + 8 * t + 4));
  }
}

extern "C" void kernel_launch(void* const* d_in, const int* in_sizes, int n_in,
                              void* d_out, int out_size, void* d_ws,
                              size_t ws_size, hipStream_t stream) {
  const int*   indices = (const int*)d_in[0];   // [4, 4096] int32
  const int*   weight  = (const int*)d_in[1];   // [128000, 2048] int32
  const float* scales  = (const float*)d_in[2]; // [128000, 128] f32
  float*       out     = (float*)d_out;         // [4, 4096, 4096] f32

  const int ntok = in_sizes[0];  // 16384 token slots
  const int grid = (ntok + QE_ROWS_PER_WG - 1) / QE_ROWS_PER_WG;

  hipLaunchKernelGGL(qembed_dequant_kernel, dim3(grid), dim3(QE_THREADS), 0,
                     stream, indices, weight, scales, out, ntok);
}